// ConfGenerator_73667279061351
// MI455X (gfx1250) — compile-verified
//
#include <hip/hip_runtime.h>
#include <math.h>

#define N_NODES 50000
#define N_EDGES 800000
#define NH 256
#define EH 64
#define EPSV 0.01f

// node kernel: 12 col-tiles (8 for Wt1's 128 cols, 4 for [Ws|Wd]'s 64 cols)
#define NTILES_N 12
#define KSTEPS_N 64          // K = 256 / 4
#define KPAIRS_N (KSTEPS_N / 2)
// edge kernel: 2 col-tiles (We's 32 cols)
#define NTILES_E 2
#define KSTEPS_E 16          // K = 64 / 4
#define KPAIRS_E (KSTEPS_E / 2)

#define PACKN_FLOATS (KPAIRS_N * NTILES_N * 32 * 4)   // 49152 floats
#define PACKE_FLOATS (KPAIRS_E * NTILES_E * 32 * 4)   // 2048 floats

typedef float v2f __attribute__((ext_vector_type(2)));
typedef float v4f __attribute__((ext_vector_type(4)));
typedef float v8f __attribute__((ext_vector_type(8)));

__device__ __forceinline__ float leaky(float x) { return x >= 0.0f ? x : 0.01f * x; }

// D = A(16x4 f32) * B(4x16 f32) + C, fp32 accumulate. wave32; EXEC all-ones.
__device__ __forceinline__ v8f wmma_k4(v2f a, v2f b, v8f c) {
    return __builtin_amdgcn_wmma_f32_16x16x4_f32(
        /*neg_a=*/false, a, /*neg_b=*/false, b,
        /*c_mod=*/(short)0, c, /*reuse_a=*/false, /*reuse_b=*/false);
}

// B fragment for WMMA f32 16x16x4: lane l holds B[kbase + {0,1}][l%16],
// kbase = (l/16)*2, for a given col-tile. We pre-swizzle B so each lane reads
// its two consecutive-k-step fragments as one contiguous float4 (b128).

// ---------------------------------------------------------------------------
// Repack kernels: B matrices -> fragment-linear order
// pack index: ((kpair * NTILES + ct) * 32 + lane) * 4 floats
// ---------------------------------------------------------------------------
__global__ void pack_node_B_kernel(const float* __restrict__ Wt1,
                                   const float* __restrict__ Wr1,
                                   float* __restrict__ packBn)
{
    int t = blockIdx.x * blockDim.x + threadIdx.x;        // fragment-pair slot
    if (t >= KPAIRS_N * NTILES_N * 32) return;
    int lane = t & 31;
    int ct   = (t >> 5) % NTILES_N;
    int kkp  = t / (NTILES_N * 32);
    int half = lane >> 4, lm = lane & 15;

    v4f v;
    #pragma unroll
    for (int p = 0; p < 2; ++p) {
        int krow = (kkp * 2 + p) * 4 + half * 2;
        float b0, b1;
        if (ct < 8) {                                     // Wt1 (256 x 128)
            b0 = Wt1[(size_t)krow       * 128 + ct * 16 + lm];
            b1 = Wt1[(size_t)(krow + 1) * 128 + ct * 16 + lm];
        } else {                                          // [Ws | Wd] (256 x 64)
            int g = (ct - 8) * 16 + lm;
            const float* base = (g < 32) ? (Wr1 + (size_t)EH * 32 + g)
                                         : (Wr1 + (size_t)(EH + NH) * 32 + (g - 32));
            b0 = base[(size_t)krow * 32];
            b1 = base[(size_t)(krow + 1) * 32];
        }
        v[2 * p]     = b0;
        v[2 * p + 1] = b1;
    }
    *(v4f*)(packBn + (size_t)t * 4) = v;
}

__global__ void pack_edge_B_kernel(const float* __restrict__ Wr1,   // We = rows 0..63
                                   float* __restrict__ packBe)
{
    int t = blockIdx.x * blockDim.x + threadIdx.x;
    if (t >= KPAIRS_E * NTILES_E * 32) return;
    int lane = t & 31;
    int ct   = (t >> 5) % NTILES_E;
    int kkp  = t / (NTILES_E * 32);
    int half = lane >> 4, lm = lane & 15;

    v4f v;
    #pragma unroll
    for (int p = 0; p < 2; ++p) {
        int krow = (kkp * 2 + p) * 4 + half * 2;
        v[2 * p]     = Wr1[(size_t)krow       * 32 + ct * 16 + lm];
        v[2 * p + 1] = Wr1[(size_t)(krow + 1) * 32 + ct * 16 + lm];
    }
    *(v4f*)(packBe + (size_t)t * 4) = v;
}

// ---------------------------------------------------------------------------
// Kernel A: per 16-node tile (one wave32 per block)
//   acc[0..7]  : H1 = node_emb @ Wt1        (16 x 128)
//   acc[8..11] : [hs | hd] = node_emb @ Wsd (16 x 64)
//   then bias+leaky on H1, 128->3 head through LDS -> rho/phi/theta
// ---------------------------------------------------------------------------
__global__ __launch_bounds__(32)
void node_gemm_kernel(const float* __restrict__ node_emb,
                      const float* __restrict__ packBn,
                      const float* __restrict__ bt1,
                      const float* __restrict__ Wt2, const float* __restrict__ bt2,
                      float* __restrict__ hs, float* __restrict__ hd,
                      float* __restrict__ out_torsion)
{
    __shared__ float sh[16 * 128];
    const int lane = threadIdx.x;
    const int half = lane >> 4;
    const int lm   = lane & 15;
    const int node_base = blockIdx.x * 16;

    v8f z = {};
    v8f acc[NTILES_N];
    #pragma unroll
    for (int i = 0; i < NTILES_N; ++i) acc[i] = z;

    const float* arow = node_emb + (size_t)(node_base + lm) * NH + half * 2;

    #pragma unroll 2
    for (int kkp = 0; kkp < KPAIRS_N; ++kkp) {
        v2f a0 = *(const v2f*)(arow + kkp * 8);
        v2f a1 = *(const v2f*)(arow + kkp * 8 + 4);
        const float* bp = packBn + (((size_t)kkp * NTILES_N) * 32 + lane) * 4;
        #pragma unroll
        for (int ct = 0; ct < NTILES_N; ++ct) {
            v4f b = *(const v4f*)(bp + (size_t)ct * 32 * 4);   // b128, coalesced
            v2f blo; blo[0] = b[0]; blo[1] = b[1];
            v2f bhi; bhi[0] = b[2]; bhi[1] = b[3];
            acc[ct] = wmma_k4(a0, blo, acc[ct]);
            acc[ct] = wmma_k4(a1, bhi, acc[ct]);
        }
    }

    // Store hs / hd (raw pre-activation; br1 added in edge stage).
    // C layout: VGPR j, lanes 0-15 -> M=j, lanes 16-31 -> M=j+8; N = lm.
    #pragma unroll
    for (int ct = 0; ct < 4; ++ct) {
        int g = ct * 16 + lm;
        float* dst = (g < 32) ? (hs + g) : (hd + (g - 32));
        #pragma unroll
        for (int j = 0; j < 8; ++j) {
            int row = node_base + j + 8 * half;
            dst[(size_t)row * 32] = acc[8 + ct][j];
        }
    }

    // bias + leaky_relu on H1, stage through LDS for the 128->3 head
    #pragma unroll
    for (int ct = 0; ct < 8; ++ct) {
        int col = ct * 16 + lm;
        float bb = bt1[col];
        #pragma unroll
        for (int j = 0; j < 8; ++j) {
            int row = j + 8 * half;
            sh[row * 128 + col] = leaky(acc[ct][j] + bb);
        }
    }
    __syncthreads();

    if (lane < 16) {
        float t0 = bt2[0], t1 = bt2[1], t2 = bt2[2];
        const float* hrow = sh + lane * 128;
        #pragma unroll 4
        for (int k = 0; k < 128; ++k) {
            float hv = hrow[k];
            t0 += hv * Wt2[k * 3 + 0];
            t1 += hv * Wt2[k * 3 + 1];
            t2 += hv * Wt2[k * 3 + 2];
        }
        float rho = sqrtf(t0 * t0 + t1 * t1 + t2 * t2);
        float inv = 1.0f / rho;
        float n0 = t0 * inv, n1 = t1 * inv, n2 = t2 * inv;
        float theta = (fabsf(n0) > fabsf(n1))
                          ? atan2f(n1, n0)
                          : (1.57079632679489662f - atan2f(n0, n1));
        float phi = acosf(fminf(fmaxf(n2, -1.0f + EPSV), 1.0f - EPSV));
        int m = node_base + lane;
        out_torsion[(size_t)m * 3 + 0] = rho;
        out_torsion[(size_t)m * 3 + 1] = phi;
        out_torsion[(size_t)m * 3 + 2] = theta;
    }
}

// ---------------------------------------------------------------------------
// Kernel B: per 16-edge tile
//   h  = leaky(edge_emb @ We + hs[src] + hd[dest] + br1)   (16x32, WMMA)
//   ro = h @ Wr2 + br2 -> edge_len (softplus), step (sigmoid)
// ---------------------------------------------------------------------------
__global__ __launch_bounds__(32)
void edge_mlp_kernel(const float* __restrict__ edge_emb,
                     const float* __restrict__ packBe,
                     const float* __restrict__ br1,
                     const float* __restrict__ Wr2, const float* __restrict__ br2,
                     const float* __restrict__ hs, const float* __restrict__ hd,
                     const int* __restrict__ src, const int* __restrict__ dst,
                     float* __restrict__ edge_len, float* __restrict__ step_size)
{
    __shared__ float sh[16 * 32];
    const int lane = threadIdx.x;
    const int half = lane >> 4;
    const int lm   = lane & 15;
    const int e_base = blockIdx.x * 16;

    v8f z = {};
    v8f acc[NTILES_E]; acc[0] = z; acc[1] = z;

    const float* arow = edge_emb + (size_t)(e_base + lm) * EH + half * 2;
    #pragma unroll
    for (int kkp = 0; kkp < KPAIRS_E; ++kkp) {
        v2f a0 = *(const v2f*)(arow + kkp * 8);
        v2f a1 = *(const v2f*)(arow + kkp * 8 + 4);
        const float* bp = packBe + (((size_t)kkp * NTILES_E) * 32 + lane) * 4;
        #pragma unroll
        for (int ct = 0; ct < NTILES_E; ++ct) {
            v4f b = *(const v4f*)(bp + (size_t)ct * 32 * 4);
            v2f blo; blo[0] = b[0]; blo[1] = b[1];
            v2f bhi; bhi[0] = b[2]; bhi[1] = b[3];
            acc[ct] = wmma_k4(a0, blo, acc[ct]);
            acc[ct] = wmma_k4(a1, bhi, acc[ct]);
        }
    }

    int es[8], ed[8];
    #pragma unroll
    for (int j = 0; j < 8; ++j) {
        int e = e_base + j + 8 * half;
        es[j] = src[e];
        ed[j] = dst[e];
    }

    #pragma unroll
    for (int ct = 0; ct < NTILES_E; ++ct) {
        int col = ct * 16 + lm;
        float bb = br1[col];
        #pragma unroll
        for (int j = 0; j < 8; ++j) {
            float v = acc[ct][j]
                    + hs[(size_t)es[j] * 32 + col]
                    + hd[(size_t)ed[j] * 32 + col]
                    + bb;
            sh[(j + 8 * half) * 32 + col] = leaky(v);
        }
    }
    __syncthreads();

    if (lane < 16) {
        float r0 = br2[0], r1 = br2[1];
        const float* hrow = sh + lane * 32;
        #pragma unroll
        for (int k = 0; k < 32; ++k) {
            float hv = hrow[k];
            r0 += hv * Wr2[k * 2 + 0];
            r1 += hv * Wr2[k * 2 + 1];
        }
        int e = e_base + lane;
        float x  = r0 + 1.0f;
        float sp = (x > 20.0f) ? x : log1pf(expf(x));        // softplus
        edge_len[e]  = sp;
        step_size[e] = 1.0f / (1.0f + expf(-(r1 - 2.0f)));   // sigmoid
    }
}

// ---------------------------------------------------------------------------
// Refiner loop kernels (L2-resident gather / atomic scatter)
// ---------------------------------------------------------------------------
__global__ void init_kernel(const float* __restrict__ cart_init,
                            float* __restrict__ coords,
                            float* __restrict__ aggr, int n)
{
    int i = blockIdx.x * blockDim.x + threadIdx.x;
    if (i < n) { coords[i] = cart_init[i]; aggr[i] = 0.0f; }
}

__global__ void force_kernel(const float* __restrict__ coords,
                             const int* __restrict__ src, const int* __restrict__ dst,
                             const float* __restrict__ edge_len,
                             const float* __restrict__ step_size,
                             float* __restrict__ aggr)
{
    int e = blockIdx.x * blockDim.x + threadIdx.x;
    if (e >= N_EDGES) return;
    int s = src[e], d = dst[e];
    float dx = coords[d * 3 + 0] - coords[s * 3 + 0];
    float dy = coords[d * 3 + 1] - coords[s * 3 + 1];
    float dz = coords[d * 3 + 2] - coords[s * 3 + 2];
    float dist = sqrtf(dx * dx + dy * dy + dz * dz);
    float f = 2.0f * step_size[e] * (edge_len[e] - dist) / (dist + EPSV);
    atomicAdd(&aggr[d * 3 + 0], f * dx);
    atomicAdd(&aggr[d * 3 + 1], f * dy);
    atomicAdd(&aggr[d * 3 + 2], f * dz);
}

__global__ void update_kernel(float* __restrict__ coords, float* __restrict__ aggr)
{
    int n = blockIdx.x * blockDim.x + threadIdx.x;
    if (n >= N_NODES) return;
    float ax = aggr[n * 3 + 0], ay = aggr[n * 3 + 1], az = aggr[n * 3 + 2];
    float an = sqrtf(ax * ax + ay * ay + az * az) + EPSV;
    float sc = fminf(an, 0.5f) / an;
    coords[n * 3 + 0] += ax * sc;
    coords[n * 3 + 1] += ay * sc;
    coords[n * 3 + 2] += az * sc;
    aggr[n * 3 + 0] = 0.0f;   // reset for next iteration
    aggr[n * 3 + 1] = 0.0f;
    aggr[n * 3 + 2] = 0.0f;
}

// ---------------------------------------------------------------------------
extern "C" void kernel_launch(void* const* d_in, const int* in_sizes, int n_in,
                              void* d_out, int out_size, void* d_ws, size_t ws_size,
                              hipStream_t stream)
{
    const float* node_emb  = (const float*)d_in[0];
    const float* edge_emb  = (const float*)d_in[1];
    const float* cart_init = (const float*)d_in[2];
    const float* Wt1 = (const float*)d_in[3];
    const float* bt1 = (const float*)d_in[4];
    const float* Wt2 = (const float*)d_in[5];
    const float* bt2 = (const float*)d_in[6];
    const float* Wr1 = (const float*)d_in[7];
    const float* br1 = (const float*)d_in[8];
    const float* Wr2 = (const float*)d_in[9];
    const float* br2 = (const float*)d_in[10];
    const int*   ei  = (const int*)d_in[11];
    const int* src = ei;
    const int* dst = ei + N_EDGES;

    float* out     = (float*)d_out;
    float* coords  = out;                       // output 0: (50000, 3)
    float* torsion = out + (size_t)N_NODES * 3; // output 1: (50000, 3)

    // Workspace layout (floats); all chunk offsets are 16-byte aligned
    float* ws     = (float*)d_ws;
    float* hs     = ws;                               // 50000*32
    float* hd     = hs     + (size_t)N_NODES * 32;    // 50000*32
    float* elen   = hd     + (size_t)N_NODES * 32;    // 800000
    float* estep  = elen   + (size_t)N_EDGES;         // 800000
    float* aggr   = estep  + (size_t)N_EDGES;         // 50000*3
    float* packBn = aggr   + (size_t)N_NODES * 3;     // 49152
    float* packBe = packBn + (size_t)PACKN_FLOATS;    // 2048

    // One-time (per launch) B-matrix repack into WMMA fragment order
    pack_node_B_kernel<<<(KPAIRS_N * NTILES_N * 32 + 255) / 256, 256, 0, stream>>>(
        Wt1, Wr1, packBn);
    pack_edge_B_kernel<<<(KPAIRS_E * NTILES_E * 32 + 255) / 256, 256, 0, stream>>>(
        Wr1, packBe);

    // coords <- cart_init, aggr <- 0 (both 150000 floats)
    init_kernel<<<(N_NODES * 3 + 255) / 256, 256, 0, stream>>>(
        cart_init, coords, aggr, N_NODES * 3);

    // Node-side GEMMs + torsion head (WMMA)
    node_gemm_kernel<<<N_NODES / 16, 32, 0, stream>>>(
        node_emb, packBn, bt1, Wt2, bt2, hs, hd, torsion);

    // Edge MLP (WMMA) -> edge_len / step_size
    edge_mlp_kernel<<<N_EDGES / 16, 32, 0, stream>>>(
        edge_emb, packBe, br1, Wr2, br2, hs, hd, src, dst, elen, estep);

    // 10 force-directed refinement steps
    for (int it = 0; it < 10; ++it) {
        force_kernel<<<(N_EDGES + 255) / 256, 256, 0, stream>>>(
            coords, src, dst, elen, estep, aggr);
        update_kernel<<<(N_NODES + 255) / 256, 256, 0, stream>>>(coords, aggr);
    }
}